// GraphSAGE_87282325390047
// MI455X (gfx1250) — compile-verified
//
#include <hip/hip_runtime.h>
#include <stdint.h>

namespace {

constexpr int kN  = 100000;   // nodes
constexpr int kE  = 1600000;  // edges
constexpr int kF0 = 256;      // NFEAT
constexpr int kF1 = 128;      // OUTFEAT == OUTD1
constexpr int kC  = 40;       // NCLASS
constexpr float kAlpha = 0.2f;

typedef __attribute__((ext_vector_type(2))) float v2f;
typedef __attribute__((ext_vector_type(8))) float v8f;

__device__ __forceinline__ float leaky_relu(float v) {
  return v >= 0.f ? v : kAlpha * v;
}

// ---------------------------------------------------------------------------
// Stage 1: h = leaky(x @ W_lin + b_lin)
// A: [kN, 256] f32 row-major, B: [256, 128] f32 row-major, O: [kN, 128].
// One wave per 16-row stripe; wave computes all 8 column tiles (16x128) so the
// A fragment is loaded once per K-step. Uses V_WMMA_F32_16X16X4_F32.
//
// f32 WMMA fragment layout (ISA 7.12.2):
//   A (16x4):  lane L (0..15): row=L,  v.x=K0,v.y=K1 ; lane L+16: v.x=K2,v.y=K3
//   B (4x16):  lane L (0..15): col=L,  v.x=row K0, v.y=row K1; lanes 16..31: K2,K3
//   C (16x16): vgpr i: lanes 0..15 -> M=i, lanes 16..31 -> M=i+8, N=lane&15
// ---------------------------------------------------------------------------
__global__ __launch_bounds__(256) void k_lin(const float* __restrict__ A,
                                             const float* __restrict__ B,
                                             const float* __restrict__ bias,
                                             float* __restrict__ O) {
  const int lane  = threadIdx.x & 31;
  const int wave  = threadIdx.x >> 5;
  const int mtile = blockIdx.x * 8 + wave;          // 16-row tile index
  if (mtile * 16 >= kN) return;                     // wave-uniform exit
  const int m0    = mtile * 16;
  const int arow  = m0 + (lane & 15);
  const int khalf = (lane >> 4) * 2;                // 0 or 2
  const int col   = lane & 15;

  v8f acc[8] = {};

  for (int k0 = 0; k0 < kF0; k0 += 4) {
    const float* ap = A + (size_t)arow * kF0 + k0 + khalf;
    v2f a;
    a.x = ap[0];
    a.y = ap[1];
#pragma unroll
    for (int nt = 0; nt < 8; ++nt) {
      const float* bp = B + (size_t)(k0 + khalf) * kF1 + nt * 16 + col;
      v2f b;
      b.x = bp[0];
      b.y = bp[kF1];
      acc[nt] = __builtin_amdgcn_wmma_f32_16x16x4_f32(
          false, a, false, b, (short)0, acc[nt], false, false);
    }
  }

  const int rbase = m0 + (lane >> 4) * 8;
#pragma unroll
  for (int nt = 0; nt < 8; ++nt) {
    const int c  = nt * 16 + col;
    const float bv = bias[c];
#pragma unroll
    for (int i = 0; i < 8; ++i) {
      float v = acc[nt][i] + bv;
      O[(size_t)(rbase + i) * kF1 + c] = leaky_relu(v);
    }
  }
}

// ---------------------------------------------------------------------------
// Fused SAGE layer: O = act(H @ Wself + (AGG * dinv) @ Wneigh + bias)
// K = inner dim (128), NT = number of 16-col tiles, NCOLS = true col count.
// Degree-reciprocal applied as a per-lane scalar to the neighbor A fragment.
// ---------------------------------------------------------------------------
template <int K, int NT, bool LEAKY, int NCOLS>
__global__ __launch_bounds__(256) void k_sage(const float* __restrict__ H,
                                              const float* __restrict__ AGG,
                                              const float* __restrict__ DINV,
                                              const float* __restrict__ Wself,
                                              const float* __restrict__ Wneigh,
                                              const float* __restrict__ bias,
                                              float* __restrict__ O) {
  const int lane  = threadIdx.x & 31;
  const int wave  = threadIdx.x >> 5;
  const int mtile = blockIdx.x * 8 + wave;
  if (mtile * 16 >= kN) return;                     // wave-uniform exit
  const int m0    = mtile * 16;
  const int arow  = m0 + (lane & 15);
  const int khalf = (lane >> 4) * 2;
  const int col   = lane & 15;
  const float dscale = DINV[arow];                  // per-lane row scalar

  v8f acc[NT] = {};

  for (int k0 = 0; k0 < K; k0 += 4) {
    const float* hp = H   + (size_t)arow * K + k0 + khalf;
    const float* gp = AGG + (size_t)arow * K + k0 + khalf;
    v2f as, an;
    as.x = hp[0];
    as.y = hp[1];
    an.x = gp[0] * dscale;
    an.y = gp[1] * dscale;
#pragma unroll
    for (int nt = 0; nt < NT; ++nt) {
      const int c = nt * 16 + col;
      v2f bs, bn;
      if (NCOLS % 16 == 0 || c < NCOLS) {
        const float* bsp = Wself  + (size_t)(k0 + khalf) * NCOLS + c;
        const float* bnp = Wneigh + (size_t)(k0 + khalf) * NCOLS + c;
        bs.x = bsp[0];     bs.y = bsp[NCOLS];
        bn.x = bnp[0];     bn.y = bnp[NCOLS];
      } else {
        bs.x = 0.f; bs.y = 0.f; bn.x = 0.f; bn.y = 0.f;
      }
      acc[nt] = __builtin_amdgcn_wmma_f32_16x16x4_f32(
          false, as, false, bs, (short)0, acc[nt], false, false);
      acc[nt] = __builtin_amdgcn_wmma_f32_16x16x4_f32(
          false, an, false, bn, (short)0, acc[nt], false, false);
    }
  }

  const int rbase = m0 + (lane >> 4) * 8;
#pragma unroll
  for (int nt = 0; nt < NT; ++nt) {
    const int c = nt * 16 + col;
    if (NCOLS % 16 == 0 || c < NCOLS) {
      const float bv = bias[c];
#pragma unroll
      for (int i = 0; i < 8; ++i) {
        float v = acc[nt][i] + bv;
        if (LEAKY) v = leaky_relu(v);
        O[(size_t)(rbase + i) * NCOLS + c] = v;
      }
    }
  }
}

// ---------------------------------------------------------------------------
// Edge aggregation: AGG[dst] += H[src]. One wave per edge; each lane owns a
// float4 (32 lanes x 4 = 128 feats): one B128 gather + 4 L2 f32 atomics.
// ---------------------------------------------------------------------------
__global__ __launch_bounds__(256) void k_scatter(const long long* __restrict__ src,
                                                 const long long* __restrict__ dst,
                                                 const float* __restrict__ H,
                                                 float* __restrict__ AGG,
                                                 int nedges) {
  const int lane = threadIdx.x & 31;
  const int e    = (int)((blockIdx.x * (unsigned)blockDim.x + threadIdx.x) >> 5);
  if (e >= nedges) return;
  const size_t s = (size_t)src[e];
  const size_t d = (size_t)dst[e];
  const float4 v = ((const float4*)(H + s * kF1))[lane];
  float* ap = AGG + d * kF1 + lane * 4;
  atomicAdd(ap + 0, v.x);
  atomicAdd(ap + 1, v.y);
  atomicAdd(ap + 2, v.z);
  atomicAdd(ap + 3, v.w);
}

__global__ void k_deg(const long long* __restrict__ dst, float* __restrict__ deg,
                      int nedges) {
  int e = blockIdx.x * blockDim.x + threadIdx.x;
  if (e < nedges) atomicAdd(deg + (size_t)dst[e], 1.0f);
}

__global__ void k_deginv(float* __restrict__ deg, int n) {
  int i = blockIdx.x * blockDim.x + threadIdx.x;
  if (i < n) deg[i] = 1.0f / fmaxf(deg[i], 1.0f);
}

__global__ void k_zero(float* __restrict__ p, long long n) {
  long long i = (long long)blockIdx.x * blockDim.x + threadIdx.x;
  if (i < n) p[i] = 0.0f;
}

// Row-wise log-softmax over 40 classes, in place on d_out.
__global__ void k_logsoftmax(float* __restrict__ O, int n) {
  int r = blockIdx.x * blockDim.x + threadIdx.x;
  if (r >= n) return;
  float* row = O + (size_t)r * kC;
  float m = row[0];
#pragma unroll
  for (int c = 1; c < kC; ++c) m = fmaxf(m, row[c]);
  float s = 0.f;
#pragma unroll
  for (int c = 0; c < kC; ++c) s += __expf(row[c] - m);
  const float lse = m + __logf(s);
#pragma unroll
  for (int c = 0; c < kC; ++c) row[c] -= lse;
}

}  // namespace

extern "C" void kernel_launch(void* const* d_in, const int* in_sizes, int n_in,
                              void* d_out, int out_size, void* d_ws, size_t ws_size,
                              hipStream_t stream) {
  // setup_inputs order:
  // 0:x 1:adj 2:edge_index 3:W_lin 4:b_lin 5:W1_self 6:W1_neigh 7:b1
  // 8:W2_self 9:W2_neigh 10:b2
  const float*     x       = (const float*)d_in[0];
  const long long* ei      = (const long long*)d_in[2];   // int64 [2, E]
  const float*     W_lin   = (const float*)d_in[3];
  const float*     b_lin   = (const float*)d_in[4];
  const float*     W1s     = (const float*)d_in[5];
  const float*     W1n     = (const float*)d_in[6];
  const float*     b1      = (const float*)d_in[7];
  const float*     W2s     = (const float*)d_in[8];
  const float*     W2n     = (const float*)d_in[9];
  const float*     b2      = (const float*)d_in[10];
  const long long* src     = ei;
  const long long* dstv    = ei + kE;
  float*           out     = (float*)d_out;

  // Workspace layout (f32): h [N*128] | h1 [N*128] | agg [N*128] | deg [N]
  float* h   = (float*)d_ws;
  float* h1  = h   + (size_t)kN * kF1;
  float* agg = h1  + (size_t)kN * kF1;
  float* deg = agg + (size_t)kN * kF1;

  const int gemm_blocks    = (kN / 16 + 7) / 8;           // 8 waves/block
  const int scatter_blocks = (kE * 32 + 255) / 256;       // 1 wave/edge
  const long long aggN     = (long long)kN * kF1;

  // Stage 1: h = leaky(x @ W_lin + b_lin)
  k_lin<<<gemm_blocks, 256, 0, stream>>>(x, W_lin, b_lin, h);

  // Degree (shared by both SAGE layers): zero agg+deg (adjacent), count, invert.
  k_zero<<<(int)((aggN + kN + 255) / 256), 256, 0, stream>>>(agg, aggN + kN);
  k_deg<<<(kE + 255) / 256, 256, 0, stream>>>(dstv, deg, kE);
  k_deginv<<<(kN + 255) / 256, 256, 0, stream>>>(deg, kN);

  // SAGE layer 1
  k_scatter<<<scatter_blocks, 256, 0, stream>>>(src, dstv, h, agg, kE);
  k_sage<kF1, 8, true, kF1><<<gemm_blocks, 256, 0, stream>>>(
      h, agg, deg, W1s, W1n, b1, h1);

  // SAGE layer 2 (re-zero agg, re-aggregate, GEMM to 40 classes padded to 48)
  k_zero<<<(int)((aggN + 255) / 256), 256, 0, stream>>>(agg, aggN);
  k_scatter<<<scatter_blocks, 256, 0, stream>>>(src, dstv, h1, agg, kE);
  k_sage<kF1, 3, false, kC><<<gemm_blocks, 256, 0, stream>>>(
      h1, agg, deg, W2s, W2n, b2, out);

  // log-softmax in place on d_out
  k_logsoftmax<<<(kN + 255) / 256, 256, 0, stream>>>(out, kN);
}